// BiARMA_82480551952879
// MI455X (gfx1250) — compile-verified
//
#include <hip/hip_runtime.h>
#include <hip/hip_bf16.h>

typedef __attribute__((ext_vector_type(2))) float v2f;
typedef __attribute__((ext_vector_type(8))) float v8f;

#define N_NODES 100000
#define N_EDGES 1000000
#define IN_CH   128
#define HID_CH  64
#define N_CLS   40

// ---------------------------------------------------------------- utilities
__global__ void zero_f(float* __restrict__ p, int n) {
  int i = blockIdx.x * blockDim.x + threadIdx.x;
  if (i < n) p[i] = 0.0f;
}

__global__ void degree_kernel(const int* __restrict__ col, float* __restrict__ deg, int nE) {
  int i = blockIdx.x * blockDim.x + threadIdx.x;
  if (i < nE) atomicAdd(&deg[col[i]], 1.0f);
}

__global__ void dinv_kernel(float* __restrict__ d, int n) {
  int i = blockIdx.x * blockDim.x + threadIdx.x;
  if (i < n) {
    float v = d[i];
    d[i] = (v > 0.0f) ? rsqrtf(fmaxf(v, 1e-12f)) : 0.0f;
  }
}

// ------------------------------------------------- edge gather / scatter-add
// thread = (edge, 4-float chunk); consecutive threads share an edge -> row/col
// loads broadcast, h-row reads coalesce. Tables are L2-resident (<=25.6MB).
template <int F, int CHUNKS>
__global__ void edge_scatter(const int* __restrict__ row, const int* __restrict__ col,
                             const float* __restrict__ dinv, const float* __restrict__ h,
                             float* __restrict__ agg, int nTotal) {
  int idx = blockIdx.x * blockDim.x + threadIdx.x;
  if (idx >= nTotal) return;
  int e = idx / CHUNKS;
  int c = idx - e * CHUNKS;
  int r = row[e];
  int t = col[e];
  float nm = dinv[r] * dinv[t];
  const float4 v = *reinterpret_cast<const float4*>(h + (size_t)r * F + c * 4);
  float* dst = agg + (size_t)t * F + c * 4;
  atomicAdd(dst + 0, nm * v.x);
  atomicAdd(dst + 1, nm * v.y);
  atomicAdd(dst + 2, nm * v.z);
  atomicAdd(dst + 3, nm * v.w);
}

// ------------------------------------------------------- f32 WMMA GEMM
// out[m][n] = (FUSE ? relu(agg[m][n] + sum_k A[m][k]W[k][n] + bias[n])
//                   : sum_k A[m][k]W[k][n])
// One wave per 16x16 tile, V_WMMA_F32_16X16X4_F32 K-steps of 4.
// W is staged zero-padded in LDS (no per-lane guards in the K loop);
// the 16xK A tile is staged in LDS and shared by all NT waves of the block.
// blockDim.x = 32*NT, grid.x = M/16 (M % 16 == 0).
template <int KDIM, int NREAL, bool FUSE>
__global__ void wmma_gemm_kernel(const float* __restrict__ A, const float* __restrict__ W,
                                 const float* __restrict__ bias, const float* __restrict__ agg,
                                 float* __restrict__ out, int M) {
  constexpr int NT = (NREAL + 15) / 16;     // n-tiles (waves per block)
  constexpr int NP = NT * 16;               // padded N
  constexpr int SW_STRIDE = NP + 8;         // conflict-free row stride for W
  constexpr int SA_STRIDE = KDIM + 4;       // conflict-free row stride for A
  constexpr int NTHREADS = 32 * NT;

  __shared__ float sW[KDIM * SW_STRIDE];
  __shared__ float sA[16 * SA_STRIDE];

  const int tid  = threadIdx.x;
  const int wave = tid >> 5;
  const int lane = tid & 31;
  const int half = lane >> 4;               // 0: K pair (k,k+1), 1: (k+2,k+3)
  const int l    = lane & 15;
  const int mBase = blockIdx.x << 4;
  const int n = (wave << 4) + l;            // padded output column (< NP)
  const bool nOK = (n < NREAL);

  // stage W (zero-padded columns) into LDS
  for (int i = tid; i < KDIM * NP; i += NTHREADS) {
    int k = i / NP;
    int nn = i - k * NP;
    sW[k * SW_STRIDE + nn] = (nn < NREAL) ? W[(size_t)k * NREAL + nn] : 0.0f;
  }
  // stage the 16 x KDIM A tile into LDS (coalesced: consecutive tid -> consecutive k)
  for (int i = tid; i < 16 * KDIM; i += NTHREADS) {
    int m = i / KDIM;
    int k = i - m * KDIM;
    sA[m * SA_STRIDE + k] = A[(size_t)(mBase + m) * KDIM + k];
  }

  // C/D layout: VGPR r -> M = mBase + r + 8*half, N = n
  v8f c = {};
  if (FUSE) {
#pragma unroll
    for (int r = 0; r < 8; ++r) {
      int m = mBase + r + 8 * half;
      c[r] = nOK ? agg[(size_t)m * NREAL + n] : 0.0f;
    }
  }

  __syncthreads();

#pragma unroll
  for (int k0 = 0; k0 < KDIM; k0 += 4) {
    const int ka = k0 + 2 * half;           // this half-wave's K pair
    v2f a, b;
    a.x = sA[l * SA_STRIDE + ka];           // A frag: lane l = row mBase+l
    a.y = sA[l * SA_STRIDE + ka + 1];
    b.x = sW[ka * SW_STRIDE + n];           // B frag: zero-padded, no guards
    b.y = sW[(ka + 1) * SW_STRIDE + n];
    c = __builtin_amdgcn_wmma_f32_16x16x4_f32(false, a, false, b, (short)0, c,
                                              false, false);
  }

#pragma unroll
  for (int r = 0; r < 8; ++r) {
    int m = mBase + r + 8 * half;
    if (nOK) {
      float v = c[r];
      if (FUSE) {
        v += bias[n];
        v = fmaxf(v, 0.0f);
      }
      out[(size_t)m * NREAL + n] = v;
    }
  }
}

// ---------------------------------------------------------------- launcher
extern "C" void kernel_launch(void* const* d_in, const int* in_sizes, int n_in,
                              void* d_out, int out_size, void* d_ws, size_t ws_size,
                              hipStream_t stream) {
  const float* x       = (const float*)d_in[0];
  const int*   eidx    = (const int*)d_in[1];
  const float* w1_init = (const float*)d_in[2];
  const float* w1_root = (const float*)d_in[3];
  const float* b1      = (const float*)d_in[4];
  const float* w2_init = (const float*)d_in[5];
  const float* w2_root = (const float*)d_in[6];
  const float* b2      = (const float*)d_in[7];
  float*       out     = (float*)d_out;

  const int* row = eidx;            // edge_index[0]
  const int* col = eidx + N_EDGES;  // edge_index[1]

  // workspace layout (floats), all 16B aligned
  float* ws   = (float*)d_ws;
  float* dinv = ws;                              // 100000  (deg, then rsqrt in place)
  float* buf1 = ws + N_NODES;                    // 100000*64 : h1, then out1
  float* buf2 = buf1 + (size_t)N_NODES * HID_CH; // 100000*64 : agg1, then h2

  const int TB = 256;
  const int mTiles = N_NODES / 16;               // 6250 exactly

  // 1) init accumulators
  zero_f<<<(N_NODES + TB - 1) / TB, TB, 0, stream>>>(dinv, N_NODES);
  zero_f<<<((N_NODES * HID_CH) + TB - 1) / TB, TB, 0, stream>>>(buf2, N_NODES * HID_CH);
  zero_f<<<((N_NODES * N_CLS) + TB - 1) / TB, TB, 0, stream>>>(out, N_NODES * N_CLS);

  // 2) gcn_norm: deg over targets, then d^-1/2
  degree_kernel<<<(N_EDGES + TB - 1) / TB, TB, 0, stream>>>(col, dinv, N_EDGES);
  dinv_kernel<<<(N_NODES + TB - 1) / TB, TB, 0, stream>>>(dinv, N_NODES);

  // 3) layer 1: h1 = x @ w1_init
  wmma_gemm_kernel<IN_CH, HID_CH, false><<<mTiles, 128, 0, stream>>>(
      x, w1_init, nullptr, nullptr, buf1, N_NODES);

  // 4) agg1[col] += norm * h1[row]
  {
    int total = N_EDGES * (HID_CH / 4);
    edge_scatter<HID_CH, HID_CH / 4><<<(total + TB - 1) / TB, TB, 0, stream>>>(
        row, col, dinv, buf1, buf2, total);
  }

  // 5) out1 = relu(agg1 + x @ w1_root + b1)   (extra BiARMA relu is idempotent)
  wmma_gemm_kernel<IN_CH, HID_CH, true><<<mTiles, 128, 0, stream>>>(
      x, w1_root, b1, buf2, buf1, N_NODES);

  // 6) layer 2: h2 = out1 @ w2_init   (reuse buf2)
  wmma_gemm_kernel<HID_CH, N_CLS, false><<<mTiles, 96, 0, stream>>>(
      buf1, w2_init, nullptr, nullptr, buf2, N_NODES);

  // 7) agg2 (in d_out) += norm * h2[row]
  {
    int total = N_EDGES * (N_CLS / 4);
    edge_scatter<N_CLS, N_CLS / 4><<<(total + TB - 1) / TB, TB, 0, stream>>>(
        row, col, dinv, buf2, out, total);
  }

  // 8) out = relu(agg2 + out1 @ w2_root + b2)   (in-place RMW of d_out)
  wmma_gemm_kernel<HID_CH, N_CLS, true><<<mTiles, 96, 0, stream>>>(
      buf1, w2_root, b2, out, out, N_NODES);
}